// DecoderIteration_42202348650562
// MI455X (gfx1250) — compile-verified
//
#include <hip/hip_runtime.h>

typedef __attribute__((ext_vector_type(2))) float v2f;
typedef __attribute__((ext_vector_type(8))) float v8f;

namespace {
constexpr float kBox  = 32.0f;
constexpr int   kD    = 64;            // hash grid per dim
constexpr int   kL    = 128;           // dense lattice side
constexpr int   kT    = 8;             // atom types
constexpr float kGW   = 0.3f;
constexpr int   kN    = 1 << 20;       // 1,048,576 points
constexpr int   kBins = kD * kD * kD;  // 262,144
constexpr int   kCap  = 8;
constexpr int   kLatElems  = kL * kL * kL * kT;   // 16,777,216
constexpr int   kBufElems  = kBins * kCap * 3;    // 6,291,456
constexpr int   kMaskElems = kBins * kCap;        // 2,097,152
constexpr int   kOutElems  = kLatElems + kBufElems + kMaskElems;
}

// ---------------------------------------------------------------------------
// Zero outputs + workspace counters (required every launch: graph replays
// would otherwise accumulate atomics on top of stale data).
// ---------------------------------------------------------------------------
__global__ void lh_zero_kernel(float4* __restrict__ out4, int n_out4,
                               int4* __restrict__ cnt4, int n_cnt4) {
  int i = blockIdx.x * blockDim.x + threadIdx.x;
  int stride = gridDim.x * blockDim.x;
  float4 fz = make_float4(0.f, 0.f, 0.f, 0.f);
  for (int k = i; k < n_out4; k += stride) out4[k] = fz;
  int4 iz = make_int4(0, 0, 0, 0);
  for (int k = i; k < n_cnt4; k += stride) cnt4[k] = iz;
}

// ---------------------------------------------------------------------------
// Spatial hash: bin -> slot via atomic counter, store point + mask.
// ---------------------------------------------------------------------------
__global__ void lh_hash_kernel(const float* __restrict__ pts,
                               const unsigned char* __restrict__ mask,
                               int* __restrict__ counts,
                               float* __restrict__ buf,
                               float* __restrict__ bufmask) {
  int i = blockIdx.x * blockDim.x + threadIdx.x;
  int stride = gridDim.x * blockDim.x;
  const float s = (float)kD / kBox;
  for (; i < kN; i += stride) {
    if (!mask[i]) continue;
    float px = pts[3 * i + 0];
    float py = pts[3 * i + 1];
    float pz = pts[3 * i + 2];
    int cx = min(kD - 1, max(0, (int)floorf(px * s)));
    int cy = min(kD - 1, max(0, (int)floorf(py * s)));
    int cz = min(kD - 1, max(0, (int)floorf(pz * s)));
    int bin = (cx * kD + cy) * kD + cz;
    int slot = atomicAdd(&counts[bin], 1);
    if (slot < kCap) {
      int base = bin * kCap + slot;
      buf[base * 3 + 0] = px;
      buf[base * 3 + 1] = py;
      buf[base * 3 + 2] = pz;
      bufmask[base] = 1.0f;
    }
  }
}

// ---------------------------------------------------------------------------
// Gaussian splat. Per 16-point batch per wave:
//   d2[m,j] = |f_m|^2 + ( -2h * f_m . (o_j+0.5) + h^2*|o_j+0.5|^2 )
// The parenthesized term is a 16x4 * 4x16 f32 matmul -> V_WMMA_F32_16X16X4_F32
// (two calls cover the 27 offsets). A-layout per ISA 32-bit 16x4 striping:
// VGPR0 = {K0 | K2}, VGPR1 = {K1 | K3} split at lane 16; B mirrored.
// D-layout: lane n holds column n, rows 0-7 (lanes 16-31: rows 8-15).
// ---------------------------------------------------------------------------
__global__ void lh_splat_kernel(const float* __restrict__ pts,
                                const unsigned char* __restrict__ mask,
                                const int* __restrict__ types,
                                float* __restrict__ lat) {
  const float h = kBox / (float)kL;           // 0.25
  const float invh = (float)kL / kBox;        // 4.0
  const float inv2s2 = 1.0f / (2.0f * kGW * kGW);

  const int lane = threadIdx.x & 31;
  const bool hi = lane >= 16;
  const int j = lane & 15;                    // WMMA column / point-in-batch id
  const int wave = (blockIdx.x * blockDim.x + threadIdx.x) >> 5;
  const int nwaves = (gridDim.x * blockDim.x) >> 5;

  // Precompute B columns + integer offsets for both offset batches.
  v2f Bmat[2];
  int oxs[2], oys[2], ozs[2];
  bool ovalid[2];
  for (int t = 0; t < 2; ++t) {
    int oidx = t * 16 + j;
    float c0 = 0.f, c1 = 0.f, c2 = 0.f, c3 = 0.f;
    int ox = 0, oy = 0, oz = 0;
    bool v = oidx < 27;
    if (v) {
      ox = oidx / 9 - 1;
      oy = (oidx / 3) % 3 - 1;
      oz = oidx % 3 - 1;
      float tx = (float)ox + 0.5f, ty = (float)oy + 0.5f, tz = (float)oz + 0.5f;
      c0 = -2.f * h * tx;
      c1 = -2.f * h * ty;
      c2 = -2.f * h * tz;
      c3 = h * h * (tx * tx + ty * ty + tz * tz);
    }
    Bmat[t].x = hi ? c2 : c0;  // K2 : K0
    Bmat[t].y = hi ? c3 : c1;  // K3 : K1
    oxs[t] = ox; oys[t] = oy; ozs[t] = oz; ovalid[t] = v;
  }

  const int nbatches = kN / 16;
  for (int b = wave; b < nbatches; b += nwaves) {
    int p = b * 16 + j;  // lane L and L+16 both mirror point (L&15)
    float px = pts[3 * p + 0];
    float py = pts[3 * p + 1];
    float pz = pts[3 * p + 2];
    int cx = min(kL - 1, max(0, (int)floorf(px * invh)));
    int cy = min(kL - 1, max(0, (int)floorf(py * invh)));
    int cz = min(kL - 1, max(0, (int)floorf(pz * invh)));
    float fx = px - (float)cx * h;
    float fy = py - (float)cy * h;
    float fz = pz - (float)cz * h;
    float f2 = fx * fx + fy * fy + fz * fz;
    int ty_ = types[p] & (kT - 1);
    float basev = mask[p] ? __expf(-f2 * inv2s2) : 0.0f;
    unsigned packed = (unsigned)cx | ((unsigned)cy << 8) |
                      ((unsigned)cz << 16) | ((unsigned)ty_ << 24);

    // A matrix: VGPR0 = {fx | fz}, VGPR1 = {fy | 1} split at lane 16.
    v2f A;
    A.x = hi ? fz : fx;
    A.y = hi ? 1.0f : fy;

    // Broadcast per-point metadata for this lane's 8 rows while EXEC is
    // all-ones (WMMA + shuffles must not sit under divergent control flow).
    unsigned pk[8];
    float bs[8];
#pragma unroll
    for (int r = 0; r < 8; ++r) {
      int src = r + (hi ? 8 : 0);
      pk[r] = __shfl(packed, src, 32);
      bs[r] = __shfl(basev, src, 32);
    }

#pragma unroll
    for (int t = 0; t < 2; ++t) {
      v8f acc = {};
      acc = __builtin_amdgcn_wmma_f32_16x16x4_f32(
          /*neg_a=*/false, A, /*neg_b=*/false, Bmat[t],
          /*c_mod=*/(short)0, acc, /*reuse_a=*/false, /*reuse_b=*/false);
      if (ovalid[t]) {
        int ox = oxs[t], oy = oys[t], oz = ozs[t];
#pragma unroll
        for (int r = 0; r < 8; ++r) {
          float base_r = bs[r];
          if (base_r == 0.0f) continue;
          float w = base_r * __expf(-acc[r] * inv2s2);
          int ccx = (int)(pk[r] & 0xFF) + ox;
          int ccy = (int)((pk[r] >> 8) & 0xFF) + oy;
          int ccz = (int)((pk[r] >> 16) & 0xFF) + oz;
          int tt = (int)(pk[r] >> 24);
          if ((unsigned)ccx < (unsigned)kL && (unsigned)ccy < (unsigned)kL &&
              (unsigned)ccz < (unsigned)kL) {
            int flat = (((ccx * kL) + ccy) * kL + ccz) * kT + tt;
            __hip_atomic_fetch_add(&lat[flat], w, __ATOMIC_RELAXED,
                                   __HIP_MEMORY_SCOPE_AGENT);
          }
        }
      }
    }
  }
}

// ---------------------------------------------------------------------------
extern "C" void kernel_launch(void* const* d_in, const int* in_sizes, int n_in,
                              void* d_out, int out_size, void* d_ws,
                              size_t ws_size, hipStream_t stream) {
  const float* pts = (const float*)d_in[0];
  const unsigned char* mask = (const unsigned char*)d_in[1];
  const int* types = (const int*)d_in[2];

  float* out = (float*)d_out;
  float* lat = out;                              // [128,128,128,8]
  float* buf = out + kLatElems;                  // [262144, 8, 3]
  float* bufmask = out + kLatElems + kBufElems;  // [262144, 8]
  int* counts = (int*)d_ws;                      // [262144] bin counters

  (void)in_sizes; (void)n_in; (void)out_size; (void)ws_size;

  // 1) Zero outputs + counters.
  lh_zero_kernel<<<2048, 256, 0, stream>>>((float4*)d_out, kOutElems / 4,
                                           (int4*)counts, kBins / 4);
  // 2) Spatial hash scatter.
  lh_hash_kernel<<<4096, 256, 0, stream>>>(pts, mask, counts, buf, bufmask);
  // 3) WMMA-assisted Gaussian splat scatter.
  lh_splat_kernel<<<2048, 256, 0, stream>>>(pts, mask, types, lat);
}